// MLoss_71562745086449
// MI455X (gfx1250) — compile-verified
//
#include <hip/hip_runtime.h>

// CDNA5 / gfx1250, wave32.
// Loss = sum_{selected cells} |y - x|^2  +  0.1 * sum_{unselected cells} x[...,0]^2
// selected := y[cell][0] > 0.5
//
// Bandwidth-bound: 128 MiB streamed -> ~5.8 us floor at 23.3 TB/s.
// Cross-lane wave reduction done with V_WMMA_F32_16X16X4_F32 (fp32-exact):
//   A (16x4, 2 VGPRs) = the wave's 64 partial sums, B (4x16) = ones,
//   D[m][n] = rowsum(A[m]); sum of 8 D regs + shfl_xor(16) = full wave sum.

typedef float v2f __attribute__((ext_vector_type(2)));
typedef float v8f __attribute__((ext_vector_type(8)));

#define THRESH 0.5f
#define ALPHA  0.1f

__device__ __forceinline__ float wave_sum_wmma(float a0, float a1) {
    v2f A; A[0] = a0;   A[1] = a1;     // A-matrix 16x4 f32: 64 partials across the wave
    v2f B; B[0] = 1.0f; B[1] = 1.0f;   // B-matrix 4x16 f32: all ones
    v8f C = {0.f, 0.f, 0.f, 0.f, 0.f, 0.f, 0.f, 0.f};
    // 8 args: (neg_a, A, neg_b, B, c_mod, C, reuse_a, reuse_b)
    v8f D = __builtin_amdgcn_wmma_f32_16x16x4_f32(
        false, A, false, B, (short)0, C, false, false);
    // lanes 0-15 hold rowsums of rows 0-7 in D[0..7]; lanes 16-31 rows 8-15.
    float s = ((D[0] + D[1]) + (D[2] + D[3])) + ((D[4] + D[5]) + (D[6] + D[7]));
    s += __shfl_xor(s, 16, 32);        // combine both halves -> full wave sum
    return s;
}

__device__ __forceinline__ float block_sum(float a0, float a1) {
    __shared__ float smem[32];
    const int lane = threadIdx.x & 31;
    const int warp = threadIdx.x >> 5;
    float w = wave_sum_wmma(a0, a1);
    if (lane == 0) smem[warp] = w;
    __syncthreads();
    float total = 0.0f;
    if (threadIdx.x == 0) {
        const int nw = (blockDim.x + 31) >> 5;
        for (int j = 0; j < nw; ++j) total += smem[j];  // fixed order: deterministic
    }
    return total;  // valid on thread 0 only
}

__global__ void mloss_stage1(const float4* __restrict__ x4,
                             const float4* __restrict__ y4,
                             float* __restrict__ partial,
                             long nv4) {
    const long stride = (long)gridDim.x * blockDim.x;     // multiple of 4
    const long tid    = (long)blockIdx.x * blockDim.x + threadIdx.x;
    const int  lane   = threadIdx.x & 31;
    const int  src    = lane & ~3;                        // group-of-4 leader (chunk 0 of cell)
    const bool isC0   = (threadIdx.x & 3) == 0;

    float acc0 = 0.0f, acc1 = 0.0f;
    for (long i = tid; i < nv4; i += 2 * stride) {
        {
            float4 xv = x4[i];
            float4 yv = y4[i];
            float  y0 = __shfl(yv.x, src, 32);            // broadcast cell objectness
            float d0 = yv.x - xv.x, d1 = yv.y - xv.y;
            float d2 = yv.z - xv.z, d3 = yv.w - xv.w;
            float sq = (d0 * d0 + d1 * d1) + (d2 * d2 + d3 * d3);
            float bg = isC0 ? ALPHA * xv.x * xv.x : 0.0f;
            acc0 += (y0 > THRESH) ? sq : bg;
        }
        long j = i + stride;
        if (j < nv4) {
            float4 xv = x4[j];
            float4 yv = y4[j];
            float  y0 = __shfl(yv.x, src, 32);
            float d0 = yv.x - xv.x, d1 = yv.y - xv.y;
            float d2 = yv.z - xv.z, d3 = yv.w - xv.w;
            float sq = (d0 * d0 + d1 * d1) + (d2 * d2 + d3 * d3);
            float bg = isC0 ? ALPHA * xv.x * xv.x : 0.0f;
            acc1 += (y0 > THRESH) ? sq : bg;
        }
    }

    float t = block_sum(acc0, acc1);
    if (threadIdx.x == 0) partial[blockIdx.x] = t;
}

__global__ void mloss_stage2(const float* __restrict__ partial,
                             int nb,
                             float* __restrict__ out) {
    float v = 0.0f;
    for (int i = threadIdx.x; i < nb; i += blockDim.x) v += partial[i];  // fixed order
    float t = block_sum(v, 0.0f);
    if (threadIdx.x == 0) out[0] = t;
}

extern "C" void kernel_launch(void* const* d_in, const int* in_sizes, int n_in,
                              void* d_out, int out_size, void* d_ws, size_t ws_size,
                              hipStream_t stream) {
    (void)n_in; (void)out_size;
    const float4* x4 = (const float4*)d_in[0];
    const float4* y4 = (const float4*)d_in[1];
    float* out     = (float*)d_out;
    float* partial = (float*)d_ws;

    const long nv4 = (long)in_sizes[0] / 4;   // 16 channels -> multiple of 4

    int nb = 2048;                            // 2048 blocks x 256 thr = 524288 lanes
    size_t maxb = ws_size / sizeof(float);
    if ((size_t)nb > maxb) nb = (int)maxb;
    if (nb < 1) nb = 1;

    mloss_stage1<<<nb, 256, 0, stream>>>(x4, y4, partial, nv4);
    mloss_stage2<<<1, 256, 0, stream>>>(partial, nb, out);
}